// AdaConv_66116726555084
// MI455X (gfx1250) — compile-verified
//
#include <hip/hip_runtime.h>

// ---------------------------------------------------------------------------
// AdaConv fused pipeline for MI455X (gfx1250, wave32)
//   N=16, CH=512, P=8, NG=PG=64, H=W=64
// Algebraic collapse: depthwise weights are broadcast over output channels ->
// all 8 channels in a group are identical after the grouped conv, and the
// pointwise conv collapses to a scalar S[n] = sum_j pk[n,j]:
//   out[n,c,h,w] = (x - mean[n,c]) * rstd[n,c]
//                  * leaky( D[n, c>>3, h, w] * S[n] + bias[n,c] )
// Bandwidth-bound: ~268 MB minimum traffic -> fused one-pass main kernel with
// GLOBAL_LOAD_ASYNC_TO_LDS_B128 tile staging (ASYNCcnt) + NT streaming stores.
// WMMA (V_WMMA_F32_16X16X4_F32) used for the bias GEMM [16,512]x[512,512]^T.
// ---------------------------------------------------------------------------

#define NB   16
#define CHN  512
#define PP   8
#define NGRP 64
#define HWSZ 4096   // 64*64

typedef __attribute__((ext_vector_type(2))) float v2f;
typedef __attribute__((ext_vector_type(8))) float v8f;

__device__ __forceinline__ float leakyf(float x) { return x >= 0.0f ? x : 0.01f * x; }

__device__ __forceinline__ float wave_sum(float v) {
#pragma unroll
    for (int off = 16; off > 0; off >>= 1) v += __shfl_xor(v, off, 32);
    return v;
}

__device__ __forceinline__ int refl64(int i) { return i < 0 ? -i : (i > 63 ? 126 - i : i); }

// CDNA5 async HBM->LDS staging (no VGPR round-trip, tracked by ASYNCcnt).
// lds_off: allocation-relative LDS byte address (low 32 bits of generic ptr).
__device__ __forceinline__ void async_load_b128(unsigned lds_off, const float* gaddr) {
    asm volatile("global_load_async_to_lds_b128 %0, %1, off"
                 :: "v"(lds_off), "v"(gaddr) : "memory");
}
__device__ __forceinline__ void wait_async0() {
    asm volatile("s_wait_asynccnt 0x0" ::: "memory");
}

// ---------------------------------------------------------------------------
// K1: s_d[n][c] = mean over the 4x4 style map  (8192 outputs)
// ---------------------------------------------------------------------------
__global__ void style_mean_kernel(const float* __restrict__ style,
                                  float* __restrict__ s_d) {
    int idx = blockIdx.x * blockDim.x + threadIdx.x;   // (n*512 + c)
    if (idx >= NB * CHN) return;
    const float* p = style + (size_t)idx * 16;
    float s = 0.f;
#pragma unroll
    for (int i = 0; i < 16; ++i) s += p[i];
    s_d[idx] = s * (1.0f / 16.0f);
}

// ---------------------------------------------------------------------------
// K2: predictor depthwise kernels
//   d[n][p][i][j] = leaky( sum_c sum_{2x2} style[n,c,i+u,j+v]*dw_w[p,c,u,v] + dw_b[p] )
// one wave per output (16*8*9 = 1152 waves), shfl reduction over c
// ---------------------------------------------------------------------------
__global__ void dw_pred_kernel(const float* __restrict__ style,
                               const float* __restrict__ dw_w,
                               const float* __restrict__ dw_b,
                               float* __restrict__ dpred) {
    int gt   = blockIdx.x * blockDim.x + threadIdx.x;
    int wave = gt >> 5;
    int lane = gt & 31;
    if (wave >= NB * PP * 9) return;
    int n   = wave / (PP * 9);
    int rem = wave % (PP * 9);
    int p   = rem / 9;
    int ij  = rem % 9;
    int i   = ij / 3, j = ij % 3;

    float acc = 0.f;
    for (int c = lane; c < CHN; c += 32) {
        const float* sp = style + ((size_t)(n * CHN + c)) * 16;
        const float* wp = dw_w + ((size_t)(p * CHN + c)) * 4;
#pragma unroll
        for (int u = 0; u < 2; ++u)
#pragma unroll
            for (int v = 0; v < 2; ++v)
                acc += sp[(i + u) * 4 + (j + v)] * wp[u * 2 + v];
    }
    acc = wave_sum(acc);
    if (lane == 0) dpred[wave] = leakyf(acc + dw_b[p]);
}

// ---------------------------------------------------------------------------
// K3: pk[n][j] = dot(s_d[n,:], pk_w[j,:]) + pk_b[j]   (128 waves)
// ---------------------------------------------------------------------------
__global__ void pk_kernel(const float* __restrict__ s_d,
                          const float* __restrict__ pk_w,
                          const float* __restrict__ pk_b,
                          float* __restrict__ pk) {
    int gt   = blockIdx.x * blockDim.x + threadIdx.x;
    int wave = gt >> 5;
    int lane = gt & 31;
    if (wave >= NB * PP) return;
    int n = wave >> 3, j = wave & 7;
    float acc = 0.f;
    for (int c = lane; c < CHN; c += 32)
        acc += s_d[n * CHN + c] * pk_w[j * CHN + c];
    acc = wave_sum(acc);
    if (lane == 0) pk[wave] = acc + pk_b[j];
}

// ---------------------------------------------------------------------------
// K4: bias[16,512] = s_d[16,512] @ pb_w[512,512]^T + pb_b  via f32 WMMA.
// One wave per 16-column tile (32 waves). K-loop of 128 x v_wmma_f32_16x16x4_f32.
// A layout (16x4 f32): lanes 0-15 -> M=lane, K={k0,k0+1}; lanes 16-31 -> K={k0+2,k0+3}
// B layout (4x16 f32): symmetric, lane = N column, VGPR/half selects K row.
// C/D layout: VGPR r -> row r (lanes 0-15) / row r+8 (lanes 16-31), lane&15 = col.
// ---------------------------------------------------------------------------
__global__ void bias_wmma_kernel(const float* __restrict__ s_d,
                                 const float* __restrict__ pb_w,
                                 const float* __restrict__ pb_b,
                                 float* __restrict__ bias) {
    int gt   = blockIdx.x * blockDim.x + threadIdx.x;
    int wave = gt >> 5;                    // 0..31 -> 16-col tile
    if (wave >= CHN / 16) return;
    int lane = threadIdx.x & 31;
    int half = lane >> 4;                  // 0: K pair (0,1), 1: K pair (2,3)
    int l    = lane & 15;                  // row for A, column for B/C/D
    int n0   = wave * 16;

    v8f acc = {};
    for (int k0 = 0; k0 < CHN; k0 += 4) {
        v2f a, b;
        const float* ap = s_d + (size_t)l * CHN + k0 + 2 * half;         // A[l][k]
        a.x = ap[0];
        a.y = ap[1];
        const float* bp = pb_w + (size_t)(n0 + l) * CHN + k0 + 2 * half; // B[k][col]=pb_w[col][k]
        b.x = bp[0];
        b.y = bp[1];
        acc = __builtin_amdgcn_wmma_f32_16x16x4_f32(
            /*neg_a=*/false, a, /*neg_b=*/false, b,
            /*c_mod=*/(short)0, acc, /*reuse_a=*/false, /*reuse_b=*/false);
    }

    float pbb = pb_b[n0 + l];
#pragma unroll
    for (int r = 0; r < 8; ++r) {
        int m = r + 8 * half;
        bias[(size_t)m * CHN + n0 + l] = acc[r] + pbb;
    }
}

// ---------------------------------------------------------------------------
// K5: fused main kernel. One block per (n, group): 256 threads, 128KB LDS tile.
//  1. stage 8 channels x 64x64 content tile into LDS via async b128 loads
//  2. per-channel instance-norm stats (one wave per channel, unbiased var)
//  3. collapsed depthwise conv D (reflect pad via in-LDS index reflection)
//  4. fused pointwise-scalar + bias + leaky + normalize + NT streaming store
// ---------------------------------------------------------------------------
__global__ void __launch_bounds__(256)
fused_main_kernel(const float* __restrict__ content,
                  const float* __restrict__ dpred,
                  const float* __restrict__ pk,
                  const float* __restrict__ bias,
                  float* __restrict__ out) {
    extern __shared__ float smem[];                 // [8][64][64] = 128 KB
    __shared__ float s_kd[PP * 9];
    __shared__ float s_bias[PP];
    __shared__ float s_mean[PP];
    __shared__ float s_rstd[PP];
    __shared__ float s_S;

    const int n   = blockIdx.x >> 6;                // /NGRP
    const int g   = blockIdx.x & 63;
    const int tid = threadIdx.x;

    // --- stage tile: 8*4096 floats = 8192 x float4, async HBM->LDS ---
    const float* base = content + ((size_t)(n * CHN + g * PP)) * HWSZ;
    const unsigned lds0 = (unsigned)(uintptr_t)smem;   // allocation-relative byte addr
#pragma unroll 4
    for (int q = tid; q < PP * HWSZ / 4; q += 256)     // 32 async b128 per thread
        async_load_b128(lds0 + (unsigned)q * 16u, base + (size_t)q * 4);

    // --- small parameter loads (independent of tile) ---
    if (tid < 72)                 s_kd[tid] = dpred[n * 72 + tid];
    else if (tid < 80)            s_bias[tid - 72] = bias[n * CHN + g * PP + (tid - 72)];
    else if (tid == 80) {
        float s = 0.f;
#pragma unroll
        for (int j = 0; j < PP; ++j) s += pk[n * PP + j];
        s_S = s;
    }

    wait_async0();          // own wave's async writes committed to LDS
    __syncthreads();        // all waves' tile slices visible

    // --- stats: wave w handles channel w ---
    {
        int w    = tid >> 5;
        int lane = tid & 31;
        const float* ch = smem + w * HWSZ;
        float s = 0.f, ss = 0.f;
        for (int idx = lane; idx < HWSZ; idx += 32) {
            float x = ch[idx];
            s  += x;
            ss += x * x;
        }
        s  = wave_sum(s);
        ss = wave_sum(ss);
        if (lane == 0) {
            float mean = s * (1.0f / HWSZ);
            float var  = (ss - s * mean) * (1.0f / (HWSZ - 1));   // unbiased (ddof=1)
            s_mean[w] = mean;
            s_rstd[w] = rsqrtf(var + 1e-5f);
        }
    }
    __syncthreads();

    // --- collapsed conv: D for 16 pixels per thread ---
    float Dv[16];
#pragma unroll
    for (int i = 0; i < 16; ++i) {
        int p = tid + (i << 8);
        int h = p >> 6, w = p & 63;
        int hr[3] = { refl64(h - 1), h, refl64(h + 1) };
        int wr[3] = { refl64(w - 1), w, refl64(w + 1) };
        float acc = 0.f;
#pragma unroll
        for (int j = 0; j < PP; ++j) {
            const float* chp = smem + j * HWSZ;
            const float* kd  = s_kd + j * 9;
#pragma unroll
            for (int u = 0; u < 3; ++u)
#pragma unroll
                for (int v = 0; v < 3; ++v)
                    acc += chp[hr[u] * 64 + wr[v]] * kd[u * 3 + v];
        }
        Dv[i] = acc;
    }

    // --- fused epilogue: 8 output channels, coalesced NT streaming stores ---
    const float Sv = s_S;
    for (int c = 0; c < PP; ++c) {
        const float bc = s_bias[c], mc = s_mean[c], rc = s_rstd[c];
        float* op = out + ((size_t)(n * CHN + g * PP + c)) * HWSZ;
        const float* xp = smem + c * HWSZ;
#pragma unroll
        for (int i = 0; i < 16; ++i) {
            int p = tid + (i << 8);
            float pred = leakyf(Dv[i] * Sv + bc);
            __builtin_nontemporal_store((xp[p] - mc) * rc * pred, op + p);
        }
    }
}

// ---------------------------------------------------------------------------
extern "C" void kernel_launch(void* const* d_in, const int* in_sizes, int n_in,
                              void* d_out, int out_size, void* d_ws, size_t ws_size,
                              hipStream_t stream) {
    const float* style   = (const float*)d_in[0];   // [16,512,4,4]
    const float* content = (const float*)d_in[1];   // [16,512,64,64]
    const float* dw_w    = (const float*)d_in[2];   // [8,512,2,2]
    const float* dw_b    = (const float*)d_in[3];   // [8]
    const float* pk_w    = (const float*)d_in[4];   // [8,512]
    const float* pk_b    = (const float*)d_in[5];   // [8]
    const float* pb_w    = (const float*)d_in[6];   // [512,512]
    const float* pb_b    = (const float*)d_in[7];   // [512]
    float* out = (float*)d_out;

    float* ws    = (float*)d_ws;
    float* s_d   = ws;                 // 16*512
    float* dpred = s_d + NB * CHN;     // 16*72
    float* pk    = dpred + NB * 72;    // 16*8
    float* bias  = pk + NB * PP;       // 16*512

    // K1: style means
    style_mean_kernel<<<(NB * CHN + 255) / 256, 256, 0, stream>>>(style, s_d);

    // K2: predictor depthwise kernels (1152 waves)
    dw_pred_kernel<<<(NB * PP * 9 * 32 + 255) / 256, 256, 0, stream>>>(style, dw_w, dw_b, dpred);

    // K3: pk dots (128 waves)
    pk_kernel<<<(NB * PP * 32 + 255) / 256, 256, 0, stream>>>(s_d, pk_w, pk_b, pk);

    // K4: bias GEMM via WMMA f32 (32 waves)
    bias_wmma_kernel<<<4, 256, 0, stream>>>(s_d, pb_w, pb_b, bias);

    // K5: fused main pass (1024 blocks, 128KB dynamic LDS each)
    const int ldsBytes = PP * HWSZ * sizeof(float);   // 131072
    hipFuncSetAttribute((const void*)fused_main_kernel,
                        hipFuncAttributeMaxDynamicSharedMemorySize, ldsBytes);
    fused_main_kernel<<<NB * NGRP, 256, ldsBytes, stream>>>(content, dpred, pk, bias, out);
}